// GeoGFM_11063835754542
// MI455X (gfx1250) — compile-verified
//
#include <hip/hip_runtime.h>
#include <math.h>

// ---------------------------------------------------------------------------
// GeoGFM on MI455X (gfx1250, wave32).
//  - fp32 WMMA (v_wmma_f32_16x16x4_f32) for all GEMMs (exact-precision path)
//  - wave-per-16-row tiles; epilogues use 16-lane shfl_xor row reductions
//  - edge log-maps: one warp per edge, 2 components/lane, global f32 atomics
// ---------------------------------------------------------------------------

typedef __attribute__((ext_vector_type(2))) float v2f;
typedef __attribute__((ext_vector_type(8))) float v8f;

#define EPSF  1e-6f
#define MAXNF 10.0f
#define PIF   3.14159265358979323846f

static __device__ __forceinline__ v8f wmma4(v2f a, v2f b, v8f c) {
  // D = A(16x4,f32) * B(4x16,f32) + C(16x16,f32)
  return __builtin_amdgcn_wmma_f32_16x16x4_f32(
      /*neg_a=*/false, a, /*neg_b=*/false, b,
      /*c_mod=*/(short)0, c, /*reuse_a=*/false, /*reuse_b=*/false);
}

static __device__ __forceinline__ float rsum16(float v) {
  v += __shfl_xor(v, 1, 32);
  v += __shfl_xor(v, 2, 32);
  v += __shfl_xor(v, 4, 32);
  v += __shfl_xor(v, 8, 32);
  return v;
}
static __device__ __forceinline__ float rsum32(float v) {
  v += __shfl_xor(v, 16, 32);
  return rsum16(v);
}

// ---------------------------------------------------------------------------
// Fused encoder: x[N,128] -> relu(xW_E+b), lorentz_exp0(xW_H+b), sphere_exp0(xW_S+b)
// One wave = 16 rows x 64 cols. A tile (16x128) preloaded once, reused 3x.
// ---------------------------------------------------------------------------
__global__ __launch_bounds__(256) void encoder_kernel(
    const float* __restrict__ x,
    const float* __restrict__ W_E, const float* __restrict__ b_E,
    const float* __restrict__ W_H, const float* __restrict__ b_H,
    const float* __restrict__ W_S, const float* __restrict__ b_S,
    float* __restrict__ outE, float* __restrict__ outH, float* __restrict__ outS,
    int N)
{
  const int lane = threadIdx.x & 31;
  const int wave = threadIdx.x >> 5;
  const int rowBase = blockIdx.x * 128 + wave * 16;
  if (rowBase >= N) return;
  const int m  = lane & 15;
  const int hi = lane >> 4;        // half-wave id
  const int kh = hi * 2;           // K sub-offset within each K=4 step
  const int arow = min(rowBase + m, N - 1);

  // Preload A tile slice for this lane: 32 K-steps, 2 floats each.
  const v2f* xr2 = reinterpret_cast<const v2f*>(x + (size_t)arow * 128);
  v2f a[32];
#pragma unroll
  for (int ks = 0; ks < 32; ++ks) a[ks] = xr2[ks * 2 + hi];

  // ---------------- Euclidean: relu(x@W_E + b_E) ----------------
  {
    v8f acc[4] = {};
#pragma unroll
    for (int ks = 0; ks < 32; ++ks) {
      const int k0 = ks * 4 + kh;
#pragma unroll
      for (int nt = 0; nt < 4; ++nt) {
        const int col = nt * 16 + m;
        v2f bb;
        bb.x = W_E[(k0 + 0) * 64 + col];
        bb.y = W_E[(k0 + 1) * 64 + col];
        acc[nt] = wmma4(a[ks], bb, acc[nt]);
      }
    }
#pragma unroll
    for (int nt = 0; nt < 4; ++nt) {
      const int col = nt * 16 + m;
      const float bias = b_E[col];
#pragma unroll
      for (int r = 0; r < 8; ++r) {
        const int rowo = rowBase + r + hi * 8;
        float v = fmaxf(acc[nt][r] + bias, 0.0f);
        if (rowo < N) outE[(size_t)rowo * 64 + col] = v;
      }
    }
  }

  // ---------------- Hyperbolic: lorentz_exp0(x@W_H + b_H) ----------------
  {
    v8f acc[4] = {};
#pragma unroll
    for (int ks = 0; ks < 32; ++ks) {
      const int k0 = ks * 4 + kh;
#pragma unroll
      for (int nt = 0; nt < 4; ++nt) {
        const int col = nt * 16 + m;
        v2f bb;
        bb.x = (col < 63) ? W_H[(k0 + 0) * 63 + col] : 0.0f;
        bb.y = (col < 63) ? W_H[(k0 + 1) * 63 + col] : 0.0f;
        acc[nt] = wmma4(a[ks], bb, acc[nt]);
      }
    }
    float p[4][8];
#pragma unroll
    for (int nt = 0; nt < 4; ++nt) {
      const int col = nt * 16 + m;
      const float bias = (col < 63) ? b_H[col] : 0.0f;
#pragma unroll
      for (int r = 0; r < 8; ++r) p[nt][r] = acc[nt][r] + bias;  // col 63 stays 0
    }
#pragma unroll
    for (int r = 0; r < 8; ++r) {
      float ss = 0.0f;
#pragma unroll
      for (int nt = 0; nt < 4; ++nt) ss += p[nt][r] * p[nt][r];
      ss = rsum16(ss);
      float n = fminf(fmaxf(sqrtf(ss), EPSF), MAXNF);
      const float sc = sinhf(n) / n;
      const int rowo = rowBase + r + hi * 8;
      if (rowo < N) {
#pragma unroll
        for (int nt = 0; nt < 4; ++nt) {
          const int col = nt * 16 + m;
          if (col < 63) outH[(size_t)rowo * 64 + col + 1] = sc * p[nt][r];
        }
        if (m == 0) outH[(size_t)rowo * 64] = coshf(n);
      }
    }
  }

  // ---------------- Spherical: sphere_exp0(x@W_S + b_S) ----------------
  {
    v8f acc[4] = {};
#pragma unroll
    for (int ks = 0; ks < 32; ++ks) {
      const int k0 = ks * 4 + kh;
#pragma unroll
      for (int nt = 0; nt < 4; ++nt) {
        const int col = nt * 16 + m;
        v2f bb;
        bb.x = (col < 63) ? W_S[(k0 + 0) * 63 + col] : 0.0f;
        bb.y = (col < 63) ? W_S[(k0 + 1) * 63 + col] : 0.0f;
        acc[nt] = wmma4(a[ks], bb, acc[nt]);
      }
    }
    float p[4][8];
#pragma unroll
    for (int nt = 0; nt < 4; ++nt) {
      const int col = nt * 16 + m;
      const float bias = (col < 63) ? b_S[col] : 0.0f;
#pragma unroll
      for (int r = 0; r < 8; ++r) p[nt][r] = acc[nt][r] + bias;
    }
#pragma unroll
    for (int r = 0; r < 8; ++r) {
      float ss = 0.0f;
#pragma unroll
      for (int nt = 0; nt < 4; ++nt) ss += p[nt][r] * p[nt][r];
      ss = rsum16(ss);
      float n = fminf(fmaxf(sqrtf(ss), EPSF), PIF - EPSF);
      const float sc = sinf(n) / n;
      const int rowo = rowBase + r + hi * 8;
      if (rowo < N) {
#pragma unroll
        for (int nt = 0; nt < 4; ++nt) {
          const int col = nt * 16 + m;
          if (col < 63) outS[(size_t)rowo * 64 + col + 1] = sc * p[nt][r];
        }
        if (m == 0) outS[(size_t)rowo * 64] = cosf(n);
      }
    }
  }
}

// ---------------------------------------------------------------------------
// Zero-fill scratch
// ---------------------------------------------------------------------------
__global__ void zero_kernel(float* __restrict__ p, int n) {
  int i = blockIdx.x * blockDim.x + threadIdx.x;
  if (i < n) p[i] = 0.0f;
}

// ---------------------------------------------------------------------------
// Edge aggregation: one warp per edge, 2 components per lane (64 dims / 32 lanes).
// u = log_{x[dst]}(x[src]); atomically accumulate into agg[dst], count into deg.
// ---------------------------------------------------------------------------
__global__ __launch_bounds__(256) void edge_lorentz_kernel(
    const float* __restrict__ xm, const int* __restrict__ src, const int* __restrict__ dst,
    int E, float* __restrict__ agg, float* __restrict__ deg)
{
  const int lane = threadIdx.x & 31;
  const int gw = (blockIdx.x * 256 + threadIdx.x) >> 5;
  const int nw = (gridDim.x * 256) >> 5;
  const int c0 = lane * 2;
  const float s0 = (c0 == 0) ? -1.0f : 1.0f;  // Minkowski sign on component 0
  for (int e = gw; e < E; e += nw) {
    const int s = src[e];
    const int d = dst[e];
    const v2f xv = *reinterpret_cast<const v2f*>(xm + (size_t)d * 64 + c0);
    const v2f yv = *reinterpret_cast<const v2f*>(xm + (size_t)s * 64 + c0);
    float ip = s0 * xv.x * yv.x + xv.y * yv.y;
    ip = rsum32(ip);                              // <x,y>_L
    const float alpha = fmaxf(-ip, 1.0f + EPSF);
    const float dd = acoshf(alpha);
    const float u0 = yv.x - alpha * xv.x;
    const float u1 = yv.y - alpha * xv.y;
    float uu = s0 * u0 * u0 + u1 * u1;
    uu = rsum32(uu);                              // <u,u>_L
    const float sc = dd / sqrtf(fmaxf(uu, EPSF));
    atomicAdd(agg + (size_t)d * 64 + c0 + 0, sc * u0);
    atomicAdd(agg + (size_t)d * 64 + c0 + 1, sc * u1);
    if (lane == 0) atomicAdd(deg + d, 1.0f);
  }
}

__global__ __launch_bounds__(256) void edge_sphere_kernel(
    const float* __restrict__ xm, const int* __restrict__ src, const int* __restrict__ dst,
    int E, float* __restrict__ agg, float* __restrict__ deg)
{
  const int lane = threadIdx.x & 31;
  const int gw = (blockIdx.x * 256 + threadIdx.x) >> 5;
  const int nw = (gridDim.x * 256) >> 5;
  const int c0 = lane * 2;
  for (int e = gw; e < E; e += nw) {
    const int s = src[e];
    const int d = dst[e];
    const v2f xv = *reinterpret_cast<const v2f*>(xm + (size_t)d * 64 + c0);
    const v2f yv = *reinterpret_cast<const v2f*>(xm + (size_t)s * 64 + c0);
    float ip = xv.x * yv.x + xv.y * yv.y;
    ip = rsum32(ip);
    const float c = fminf(fmaxf(ip, -1.0f + EPSF), 1.0f - EPSF);
    const float u0 = yv.x - c * xv.x;
    const float u1 = yv.y - c * xv.y;
    float uu = u0 * u0 + u1 * u1;
    uu = rsum32(uu);
    const float sc = acosf(c) / fmaxf(sqrtf(uu), EPSF);
    atomicAdd(agg + (size_t)d * 64 + c0 + 0, sc * u0);
    atomicAdd(agg + (size_t)d * 64 + c0 + 1, sc * u1);
    if (lane == 0) atomicAdd(deg + d, 1.0f);
  }
}

// ---------------------------------------------------------------------------
// Node update: h = (agg/deg) @ W + b ; h = proj_x(h) ; x = exp_x(h).
// WMMA GEMM K=64 with degree-normalization fused into the A-operand.
// ---------------------------------------------------------------------------
__global__ __launch_bounds__(256) void node_lorentz_kernel(
    float* __restrict__ xm, const float* __restrict__ agg, const float* __restrict__ deg,
    const float* __restrict__ W, const float* __restrict__ b, int N)
{
  const int lane = threadIdx.x & 31;
  const int wave = threadIdx.x >> 5;
  const int rowBase = blockIdx.x * 128 + wave * 16;
  if (rowBase >= N) return;
  const int m  = lane & 15;
  const int hi = lane >> 4;
  const int kh = hi * 2;
  const int arow = min(rowBase + m, N - 1);

  const float invd = 1.0f / fmaxf(deg[arow], 1.0f);
  const v2f* ar2 = reinterpret_cast<const v2f*>(agg + (size_t)arow * 64);
  v2f a[16];
#pragma unroll
  for (int ks = 0; ks < 16; ++ks) a[ks] = ar2[ks * 2 + hi] * invd;

  v8f acc[4] = {};
#pragma unroll
  for (int ks = 0; ks < 16; ++ks) {
    const int k0 = ks * 4 + kh;
#pragma unroll
    for (int nt = 0; nt < 4; ++nt) {
      const int col = nt * 16 + m;
      v2f bb;
      bb.x = W[(k0 + 0) * 64 + col];
      bb.y = W[(k0 + 1) * 64 + col];
      acc[nt] = wmma4(a[ks], bb, acc[nt]);
    }
  }

#pragma unroll
  for (int r = 0; r < 8; ++r) {
    const int rowo = rowBase + r + hi * 8;
    const int rowc = min(rowo, N - 1);
    float h[4], xx[4], sg[4], hp[4];
    float ip = 0.0f;
#pragma unroll
    for (int nt = 0; nt < 4; ++nt) {
      const int col = nt * 16 + m;
      xx[nt] = xm[(size_t)rowc * 64 + col];
      h[nt]  = acc[nt][r] + b[col];
      sg[nt] = (col == 0) ? -1.0f : 1.0f;
      ip += sg[nt] * xx[nt] * h[nt];
    }
    ip = rsum16(ip);                       // <x,h>_L
    float uu = 0.0f;
#pragma unroll
    for (int nt = 0; nt < 4; ++nt) {       // proj: h + <x,h>_L x
      hp[nt] = h[nt] + ip * xx[nt];
      uu += sg[nt] * hp[nt] * hp[nt];
    }
    uu = rsum16(uu);                       // <u,u>_L
    float n = fminf(sqrtf(fmaxf(uu, EPSF * EPSF)), MAXNF);
    const float ch = coshf(n);
    const float sh = sinhf(n) / n;
    if (rowo < N) {
#pragma unroll
      for (int nt = 0; nt < 4; ++nt) {
        const int col = nt * 16 + m;
        xm[(size_t)rowo * 64 + col] = ch * xx[nt] + sh * hp[nt];
      }
    }
  }
}

__global__ __launch_bounds__(256) void node_sphere_kernel(
    float* __restrict__ xm, const float* __restrict__ agg, const float* __restrict__ deg,
    const float* __restrict__ W, const float* __restrict__ b, int N)
{
  const int lane = threadIdx.x & 31;
  const int wave = threadIdx.x >> 5;
  const int rowBase = blockIdx.x * 128 + wave * 16;
  if (rowBase >= N) return;
  const int m  = lane & 15;
  const int hi = lane >> 4;
  const int kh = hi * 2;
  const int arow = min(rowBase + m, N - 1);

  const float invd = 1.0f / fmaxf(deg[arow], 1.0f);
  const v2f* ar2 = reinterpret_cast<const v2f*>(agg + (size_t)arow * 64);
  v2f a[16];
#pragma unroll
  for (int ks = 0; ks < 16; ++ks) a[ks] = ar2[ks * 2 + hi] * invd;

  v8f acc[4] = {};
#pragma unroll
  for (int ks = 0; ks < 16; ++ks) {
    const int k0 = ks * 4 + kh;
#pragma unroll
    for (int nt = 0; nt < 4; ++nt) {
      const int col = nt * 16 + m;
      v2f bb;
      bb.x = W[(k0 + 0) * 64 + col];
      bb.y = W[(k0 + 1) * 64 + col];
      acc[nt] = wmma4(a[ks], bb, acc[nt]);
    }
  }

#pragma unroll
  for (int r = 0; r < 8; ++r) {
    const int rowo = rowBase + r + hi * 8;
    const int rowc = min(rowo, N - 1);
    float h[4], xx[4], hp[4], y[4];
    float ip = 0.0f;
#pragma unroll
    for (int nt = 0; nt < 4; ++nt) {
      const int col = nt * 16 + m;
      xx[nt] = xm[(size_t)rowc * 64 + col];
      h[nt]  = acc[nt][r] + b[col];
      ip += xx[nt] * h[nt];
    }
    ip = rsum16(ip);                       // <x,h>
    float uu = 0.0f;
#pragma unroll
    for (int nt = 0; nt < 4; ++nt) {       // proj: h - <x,h> x
      hp[nt] = h[nt] - ip * xx[nt];
      uu += hp[nt] * hp[nt];
    }
    uu = rsum16(uu);
    const float n  = fmaxf(sqrtf(uu), EPSF);
    const float cn = cosf(n);
    const float sn = sinf(n) / n;
    float yn = 0.0f;
#pragma unroll
    for (int nt = 0; nt < 4; ++nt) {
      y[nt] = cn * xx[nt] + sn * hp[nt];
      yn += y[nt] * y[nt];
    }
    yn = rsum16(yn);
    const float inv = 1.0f / sqrtf(yn);    // renormalize to unit sphere
    if (rowo < N) {
#pragma unroll
      for (int nt = 0; nt < 4; ++nt) {
        const int col = nt * 16 + m;
        xm[(size_t)rowo * 64 + col] = y[nt] * inv;
      }
    }
  }
}

// ---------------------------------------------------------------------------
// Host launcher
// ---------------------------------------------------------------------------
extern "C" void kernel_launch(void* const* d_in, const int* in_sizes, int n_in,
                              void* d_out, int out_size, void* d_ws, size_t ws_size,
                              hipStream_t stream) {
  const float* x       = (const float*)d_in[0];
  const float* W_E     = (const float*)d_in[1];
  const float* b_E     = (const float*)d_in[2];
  const float* W_H     = (const float*)d_in[3];
  const float* b_H     = (const float*)d_in[4];
  const float* W_S     = (const float*)d_in[5];
  const float* b_S     = (const float*)d_in[6];
  const float* W_tree  = (const float*)d_in[7];
  const float* b_tree  = (const float*)d_in[8];
  const float* W_graph = (const float*)d_in[9];
  const float* b_graph = (const float*)d_in[10];
  const int*   e_tree  = (const int*)d_in[11];
  const int*   e_graph = (const int*)d_in[12];

  const int N  = in_sizes[0] / 128;
  const int Et = in_sizes[11] / 2;
  const int Eg = in_sizes[12] / 2;
  const int L  = in_sizes[7] / (64 * 64);

  float* outE = (float*)d_out;
  float* outH = outE + (size_t)N * 64;
  float* outS = outH + (size_t)N * 64;
  float* agg  = (float*)d_ws;          // N*64 floats
  float* deg  = agg + (size_t)N * 64;  // N floats

  const dim3 blk(256);
  const int gemmGrid = (N + 127) / 128;
  const int zeroGrid = (N * 64 + 255) / 256;
  const int zeroGridD = (N + 255) / 256;

  encoder_kernel<<<gemmGrid, blk, 0, stream>>>(x, W_E, b_E, W_H, b_H, W_S, b_S,
                                               outE, outH, outS, N);

  for (int l = 0; l < L; ++l) {
    // Hyperbolic learner on sub-trees
    zero_kernel<<<zeroGrid, blk, 0, stream>>>(agg, N * 64);
    zero_kernel<<<zeroGridD, blk, 0, stream>>>(deg, N);
    edge_lorentz_kernel<<<(Et + 7) / 8, blk, 0, stream>>>(outH, e_tree, e_tree + Et,
                                                          Et, agg, deg);
    node_lorentz_kernel<<<gemmGrid, blk, 0, stream>>>(outH, agg, deg,
                                                      W_tree + (size_t)l * 64 * 64,
                                                      b_tree + (size_t)l * 64, N);
    // Spherical learner on sub-graphs
    zero_kernel<<<zeroGrid, blk, 0, stream>>>(agg, N * 64);
    zero_kernel<<<zeroGridD, blk, 0, stream>>>(deg, N);
    edge_sphere_kernel<<<(Eg + 7) / 8, blk, 0, stream>>>(outS, e_graph, e_graph + Eg,
                                                         Eg, agg, deg);
    node_sphere_kernel<<<gemmGrid, blk, 0, stream>>>(outS, agg, deg,
                                                     W_graph + (size_t)l * 64 * 64,
                                                     b_graph + (size_t)l * 64, N);
  }
}